// ConvertParamsTEtoParams0TE_60438779789427
// MI455X (gfx1250) — compile-verified
//
#include <hip/hip_runtime.h>

typedef __attribute__((ext_vector_type(2))) float v2f;
typedef __attribute__((ext_vector_type(8))) float v8f;

#define NH 128
#define NV 256

// One workgroup per batch. 8 waves; each wave owns two 16-wide v-tiles.
// Streams wtTE1/wt1 once: fused elementwise wtTE2 store + f32 WMMA matvec
// (A = mu broadcast across M rows, K=4 chunks of h).
__global__ __launch_bounds__(256) void fused_convert_kernel(
    const float* __restrict__ bTE1,      // [B, NV]
    const float* __restrict__ wtTE1,     // [B, NH, NV]
    const float* __restrict__ muh1,      // [B, NH]
    const float* __restrict__ wt1,       // [B, NH, NV]
    const float* __restrict__ muhTE1,    // [B, NH]
    const float* __restrict__ varh1,     // [B, NH]
    const float* __restrict__ varhTE1,   // [B, NH]
    const float* __restrict__ sig2TE,    // [B]
    float* __restrict__ out_sig2,        // [B]
    float* __restrict__ out_bTE2,        // [B, NV]
    float* __restrict__ out_wtTE2)       // [B, NH, NV]
{
    __shared__ float s_mu[NH];    // muh1 row
    __shared__ float s_muTE[NH];  // muhTE1 row
    __shared__ float s_sa[NH];    // 0.5 * varhTE / sqrt(varh)  -> scales wt1
    __shared__ float s_sb[NH];    // sqrt(varh)                 -> scales wtTE1

    const int b   = blockIdx.x;
    const int tid = threadIdx.x;

    if (tid < NH) {
        const float v  = varh1[(size_t)b * NH + tid];
        const float sq = sqrtf(v);
        s_sb[tid]   = sq;
        s_sa[tid]   = 0.5f * varhTE1[(size_t)b * NH + tid] / sq;
        s_mu[tid]   = muh1[(size_t)b * NH + tid];
        s_muTE[tid] = muhTE1[(size_t)b * NH + tid];
    }
    if (tid == 0) out_sig2[b] = sig2TE[b];
    __syncthreads();

    const int wave  = tid >> 5;          // 0..7
    const int lane  = tid & 31;
    const int nlo   = lane & 15;         // column within 16-wide v-tile
    const int khalf = (lane >> 4) << 1;  // lanes 0-15 -> K{0,1}; lanes 16-31 -> K{2,3}

    const float* __restrict__ w1base = wtTE1     + (size_t)b * NH * NV;
    const float* __restrict__ w2base = wt1       + (size_t)b * NH * NV;
    float* __restrict__       wobase = out_wtTE2 + (size_t)b * NH * NV;

    #pragma unroll
    for (int t = 0; t < 2; ++t) {
        const int v0 = (wave * 2 + t) * 16;
        v8f acc = {};  // C/D accumulator (16x16 f32, rows replicated)

        for (int h0 = 0; h0 < NH; h0 += 4) {
            const int ka = h0 + khalf;               // this lane's first K row
            const size_t r0 = (size_t)ka * NV + v0 + nlo;
            const size_t r1 = r0 + NV;

            // Stream both weight tiles (B-operand layout: VGPRj lane -> row ka+j, col nlo)
            const float a0 = w1base[r0];             // wtTE1[ka,   v0+nlo]
            const float a1 = w1base[r1];             // wtTE1[ka+1, v0+nlo]
            const float c0 = w2base[r0];             // wt1[ka,   v0+nlo]
            const float c1 = w2base[r1];             // wt1[ka+1, v0+nlo]

            // Fused elementwise output: wtTE2 = sa[h]*wt1 + sb[h]*wtTE1
            const float sa0 = s_sa[ka], sa1 = s_sa[ka + 1];
            const float sb0 = s_sb[ka], sb1 = s_sb[ka + 1];
            wobase[r0] = fmaf(sa0, c0, sb0 * a0);
            wobase[r1] = fmaf(sa1, c1, sb1 * a1);

            // A-operand: mu[h0+k] broadcast across all 16 M rows
            // (A 16x4 layout: VGPR0 lanes0-15=K0 / lanes16-31=K2; VGPR1 = K1/K3)
            const v2f aMu   = { s_mu[ka],   s_mu[ka + 1]   };
            const v2f aMuTE = { s_muTE[ka], s_muTE[ka + 1] };
            const v2f bW1   = { a0, a1 };
            const v2f bW2   = { c0, c1 };

            // acc += muh x wtTE1  ;  acc += muhTE x wt1   (full f32 precision)
            acc = __builtin_amdgcn_wmma_f32_16x16x4_f32(
                false, aMu,   false, bW1, (short)0, acc, false, false);
            acc = __builtin_amdgcn_wmma_f32_16x16x4_f32(
                false, aMuTE, false, bW2, (short)0, acc, false, false);
        }

        // Every D row holds the same matvec result; row M=0 lives in VGPR0, lanes 0-15.
        if (lane < 16) {
            const size_t o = (size_t)b * NV + v0 + lane;
            out_bTE2[o] = bTE1[o] + acc[0];
        }
    }
}

extern "C" void kernel_launch(void* const* d_in, const int* in_sizes, int n_in,
                              void* d_out, int out_size, void* d_ws, size_t ws_size,
                              hipStream_t stream) {
    const float* bTE1    = (const float*)d_in[0];
    const float* wtTE1   = (const float*)d_in[1];
    const float* muh1    = (const float*)d_in[2];
    const float* wt1     = (const float*)d_in[3];
    const float* muhTE1  = (const float*)d_in[4];
    const float* varh1   = (const float*)d_in[5];
    const float* varhTE1 = (const float*)d_in[6];
    const float* sig2TE  = (const float*)d_in[7];

    const int B = in_sizes[7];  // 2048

    float* out      = (float*)d_out;
    float* out_sig2 = out;                          // [B]
    float* out_b    = out + B;                      // [B, NV]
    float* out_wt   = out + B + (size_t)B * NV;     // [B, NH, NV]

    fused_convert_kernel<<<B, 256, 0, stream>>>(
        bTE1, wtTE1, muh1, wt1, muhTE1, varh1, varhTE1, sig2TE,
        out_sig2, out_b, out_wt);
}